// Advanced_RGAT_3367254360424
// MI455X (gfx1250) — compile-verified
//
#include <hip/hip_runtime.h>
#include <hip/hip_bf16.h>
#include <math.h>

// ---------------- problem constants (match reference) ----------------
#define NN    50000
#define IND   128
#define HID   128
#define HEADS 4
#define RREL  8
#define EE    640000
#define EE2   200000
#define NCLS  8

#define MTB   80      // xw-GEMM row tile (divides 50000 exactly)
#define DECM  32      // decode row tile (divides 200000 exactly)

typedef float v2f __attribute__((ext_vector_type(2)));
typedef float v8f __attribute__((ext_vector_type(8)));

// =====================================================================
// K0a: init per-node attention state
__global__ __launch_bounds__(256) void k_init_attn(float* __restrict__ amax,
                                                   float* __restrict__ den) {
  int t = blockIdx.x * 256 + threadIdx.x;
  if (t < NN * HEADS) {
    amax[t] = -INFINITY;
    den[t]  = 0.0f;
  }
}

// K0b: init aggregation buffer to bias (sum-aggregation epilogue bias)
__global__ __launch_bounds__(256) void k_init_node(float* __restrict__ agg,
                                                   const float* __restrict__ b) {
  int t = blockIdx.x * 256 + threadIdx.x;   // N*HID threads
  agg[t] = b[t & (HID - 1)];
}

// =====================================================================
// K1: relation GEMM  xw[n,r,:] = x[n,:] @ W[r]   (K = 128, fp32 WMMA)
// grid = (N/MTB, R), block = 256 (8 waves; wave w owns cols [16w,16w+16),
// and 5 row-tiles of 16 -> B fragment reused 5x per load)
__global__ __launch_bounds__(256) void k_xw_gemm(const float* __restrict__ X,
                                                 const float* __restrict__ W,
                                                 float* __restrict__ XW) {
  const int n0   = blockIdx.x * MTB;
  const int r    = blockIdx.y;
  const int t    = threadIdx.x;
  const int lane = t & 31;
  const int wave = t >> 5;

  __shared__ float As[MTB][IND + 4];   // pad -> conflict-free column reads

  // stage A tile MTB x 128, coalesced float4 (10 sweeps of 1024 floats)
  #pragma unroll
  for (int it = 0; it < (MTB * IND) / 1024; ++it) {
    const int flat = it * 1024 + t * 4;
    const int row  = flat >> 7;
    const int col  = flat & 127;
    *(float4*)&As[row][col] =
        *(const float4*)(X + (size_t)(n0 + row) * IND + col);
  }

  const float* Wr = W + (size_t)r * IND * HID;
  // warm W_r (64KB) into the cache hierarchy: one global_prefetch per thread
  __builtin_prefetch(Wr + t * 64, 0, 3);
  __syncthreads();

  const int col0 = wave * 16;
  const int m    = lane & 15;          // A row / B,C column within tile
  const int kk   = (lane >> 4) * 2;    // K sub-offset for this half-wave
  const int bc   = col0 + m;

  v8f c[5] = {};
  #pragma unroll 4
  for (int k0 = 0; k0 < IND; k0 += 4) {
    v2f b;
    b.x = Wr[(size_t)(k0 + kk)     * HID + bc];
    b.y = Wr[(size_t)(k0 + kk + 1) * HID + bc];
    #pragma unroll
    for (int mb = 0; mb < 5; ++mb) {
      v2f a;
      a.x = As[mb * 16 + m][k0 + kk];
      a.y = As[mb * 16 + m][k0 + kk + 1];
      c[mb] = __builtin_amdgcn_wmma_f32_16x16x4_f32(false, a, false, b,
                                                    (short)0, c[mb],
                                                    false, false);
    }
  }

  // C layout: VGPR g -> M=g (lanes 0-15) / M=g+8 (lanes 16-31), N = lane&15
  const int mbase = (lane < 16) ? 0 : 8;
  #pragma unroll
  for (int mb = 0; mb < 5; ++mb) {
    #pragma unroll
    for (int g = 0; g < 8; ++g) {
      const int mm = mb * 16 + mbase + g;
      XW[((size_t)(n0 + mm) * RREL + r) * HID + col0 + m] = c[mb][g];
    }
  }
}

// =====================================================================
// K2: per-(node,relation) head logits: aq = xw.q, ak = xw.k
// one wave per row; 8 rows per block
__global__ __launch_bounds__(256) void k_rowdot(const float* __restrict__ XW,
                                                const float* __restrict__ Q,
                                                const float* __restrict__ K,
                                                float* __restrict__ AQ,
                                                float* __restrict__ AK) {
  __shared__ float sQ[HID * HEADS];
  __shared__ float sK[HID * HEADS];
  const int t = threadIdx.x;
  sQ[t] = Q[t]; sQ[t + 256] = Q[t + 256];
  sK[t] = K[t]; sK[t + 256] = K[t + 256];
  __syncthreads();

  const int lane = t & 31;
  const int wave = t >> 5;
  const long row = (long)blockIdx.x * 8 + wave;

  float4 v = *(const float4*)(XW + row * HID + lane * 4);
  float q[HEADS], k[HEADS];
  #pragma unroll
  for (int h = 0; h < HEADS; ++h) {
    const int j = lane * 4;
    q[h] = v.x * sQ[(j + 0) * HEADS + h] + v.y * sQ[(j + 1) * HEADS + h]
         + v.z * sQ[(j + 2) * HEADS + h] + v.w * sQ[(j + 3) * HEADS + h];
    k[h] = v.x * sK[(j + 0) * HEADS + h] + v.y * sK[(j + 1) * HEADS + h]
         + v.z * sK[(j + 2) * HEADS + h] + v.w * sK[(j + 3) * HEADS + h];
  }
  #pragma unroll
  for (int off = 16; off > 0; off >>= 1) {
    #pragma unroll
    for (int h = 0; h < HEADS; ++h) {
      q[h] += __shfl_xor(q[h], off);
      k[h] += __shfl_xor(k[h], off);
    }
  }
  if (lane < HEADS) {
    AQ[row * HEADS + lane] = q[lane];
    AK[row * HEADS + lane] = k[lane];
  }
}

// =====================================================================
// K3: attention logit + running segment max (per dst, per head)
__global__ __launch_bounds__(256) void k_edge_alpha(const float* __restrict__ AQ,
                                                    const float* __restrict__ AK,
                                                    const int* __restrict__ EI,
                                                    const int* __restrict__ ET,
                                                    float* __restrict__ ALPHA,
                                                    float* __restrict__ AMAX) {
  const int t = blockIdx.x * 256 + threadIdx.x;   // E*HEADS threads
  const int e = t >> 2, h = t & 3;
  const int s = EI[e], d = EI[EE + e], r = ET[e];
  float a = AQ[((size_t)d * RREL + r) * HEADS + h] +
            AK[((size_t)s * RREL + r) * HEADS + h];
  a = (a > 0.0f) ? a : 0.2f * a;                  // leaky_relu(0.2)
  ALPHA[t] = a;
  atomicMax(&AMAX[d * HEADS + h], a);             // GLOBAL_ATOMIC_MAX_NUM_F32
}

// K4: exp(alpha - max) + segment denominator
__global__ __launch_bounds__(256) void k_edge_exp(float* __restrict__ ALPHA,
                                                  const float* __restrict__ AMAX,
                                                  const int* __restrict__ EI,
                                                  float* __restrict__ DEN) {
  const int t = blockIdx.x * 256 + threadIdx.x;
  const int e = t >> 2, h = t & 3;
  const int d = EI[EE + e];
  const float ex = __expf(ALPHA[t] - AMAX[d * HEADS + h]);
  ALPHA[t] = ex;
  atomicAdd(&DEN[d * HEADS + h], ex);
}

// K5: attention-weighted message scatter-add (wave per edge, float4/lane)
__global__ __launch_bounds__(256) void k_edge_agg(const float* __restrict__ XW,
                                                  const float* __restrict__ EX,
                                                  const float* __restrict__ DEN,
                                                  const int* __restrict__ EI,
                                                  const int* __restrict__ ET,
                                                  float* __restrict__ AGG) {
  const int t    = blockIdx.x * 256 + threadIdx.x;
  const int e    = t >> 5;
  const int lane = t & 31;
  const int col  = lane * 4;
  const int h    = lane >> 3;                     // 32 cols per head
  const int s = EI[e], d = EI[EE + e], r = ET[e];
  const float attn = EX[e * HEADS + h] / (DEN[d * HEADS + h] + 1e-16f);
  const float4 xj = *(const float4*)(XW + ((size_t)s * RREL + r) * HID + col);
  float* o = AGG + (size_t)d * HID + col;
  atomicAdd(o + 0, xj.x * attn);
  atomicAdd(o + 1, xj.y * attn);
  atomicAdd(o + 2, xj.z * attn);
  atomicAdd(o + 3, xj.w * attn);
}

// =====================================================================
// K6: relu + LayerNorm (+ optional residual) — wave per node
__global__ __launch_bounds__(256) void k_postln(const float* __restrict__ AGG,
                                                const float* __restrict__ G,
                                                const float* __restrict__ B,
                                                const float* __restrict__ PREV,
                                                float* __restrict__ OUT) {
  const int t    = blockIdx.x * 256 + threadIdx.x;
  const int n    = t >> 5;
  const int lane = t & 31;
  const int col  = lane * 4;

  float4 v = *(const float4*)(AGG + (size_t)n * HID + col);
  v.x = fmaxf(v.x, 0.0f); v.y = fmaxf(v.y, 0.0f);
  v.z = fmaxf(v.z, 0.0f); v.w = fmaxf(v.w, 0.0f);

  float s = v.x + v.y + v.z + v.w;
  #pragma unroll
  for (int off = 16; off > 0; off >>= 1) s += __shfl_xor(s, off);
  const float mean = s * (1.0f / HID);

  const float dx = v.x - mean, dy = v.y - mean, dz = v.z - mean, dw = v.w - mean;
  float ss = dx * dx + dy * dy + dz * dz + dw * dw;
  #pragma unroll
  for (int off = 16; off > 0; off >>= 1) ss += __shfl_xor(ss, off);
  const float rs = rsqrtf(ss * (1.0f / HID) + 1e-5f);

  const float4 g4 = *(const float4*)(G + col);
  const float4 b4 = *(const float4*)(B + col);
  float4 o;
  o.x = dx * rs * g4.x + b4.x;
  o.y = dy * rs * g4.y + b4.y;
  o.z = dz * rs * g4.z + b4.z;
  o.w = dw * rs * g4.w + b4.w;
  if (PREV) {
    const float4 p = *(const float4*)(PREV + (size_t)n * HID + col);
    o.x += p.x; o.y += p.y; o.z += p.z; o.w += p.w;
  }
  *(float4*)(OUT + (size_t)n * HID + col) = o;
}

// =====================================================================
// K7: decode — gather pair rows, (256->128) WMMA + bias + exact GELU,
// then fused (128->8) classifier. DECM pairs per block, 8 waves,
// 2 row-tiles per wave (B fragment reused 2x).
__global__ __launch_bounds__(256) void k_decode(const float* __restrict__ H,
                                                const int* __restrict__ EDG,
                                                const float* __restrict__ MW1,
                                                const float* __restrict__ MB1,
                                                const float* __restrict__ MW2,
                                                const float* __restrict__ MB2,
                                                float* __restrict__ OUT) {
  const int p0   = blockIdx.x * DECM;
  const int t    = threadIdx.x;
  const int lane = t & 31;
  const int wave = t >> 5;

  __shared__ float As[DECM][2 * HID + 4];
  __shared__ float Zs[DECM][HID + 4];

  // stage A: z[row] = concat(h[e0], h[e1])  -> DECM x 256
  #pragma unroll
  for (int it = 0; it < (DECM * 2 * HID) / 1024; ++it) {
    const int flat = it * 1024 + t * 4;
    const int row  = flat >> 8;
    const int col  = flat & 255;
    const int node = EDG[(p0 + row) * 2 + (col >> 7)];
    *(float4*)&As[row][col] =
        *(const float4*)(H + (size_t)node * HID + (col & 127));
  }
  // warm mw1 (128KB)
  __builtin_prefetch(MW1 + t * 128, 0, 3);
  __syncthreads();

  const int col0 = wave * 16;
  const int m    = lane & 15;
  const int kk   = (lane >> 4) * 2;
  const int bc   = col0 + m;

  v8f c[2] = {};
  #pragma unroll 8
  for (int k0 = 0; k0 < 2 * HID; k0 += 4) {
    v2f b;
    b.x = MW1[(size_t)(k0 + kk)     * HID + bc];
    b.y = MW1[(size_t)(k0 + kk + 1) * HID + bc];
    #pragma unroll
    for (int mb = 0; mb < 2; ++mb) {
      v2f a;
      a.x = As[mb * 16 + m][k0 + kk];
      a.y = As[mb * 16 + m][k0 + kk + 1];
      c[mb] = __builtin_amdgcn_wmma_f32_16x16x4_f32(false, a, false, b,
                                                    (short)0, c[mb],
                                                    false, false);
    }
  }

  const int mbase  = (lane < 16) ? 0 : 8;
  const float bias = MB1[col0 + m];
  #pragma unroll
  for (int mb = 0; mb < 2; ++mb) {
    #pragma unroll
    for (int g = 0; g < 8; ++g) {
      const float x = c[mb][g] + bias;
      Zs[mb * 16 + mbase + g][col0 + m] =
          0.5f * x * (1.0f + erff(x * 0.70710678118654752f));  // exact gelu
    }
  }
  __syncthreads();

  // fused classifier: 32 rows x 8 classes = 256 threads
  {
    const int row = t >> 3, cls = t & 7;
    float acc = MB2[cls];
    #pragma unroll 8
    for (int j = 0; j < HID; ++j) acc += Zs[row][j] * MW2[j * NCLS + cls];
    OUT[(size_t)(p0 + row) * NCLS + cls] = acc;
  }
}

// =====================================================================
extern "C" void kernel_launch(void* const* d_in, const int* in_sizes, int n_in,
                              void* d_out, int out_size, void* d_ws, size_t ws_size,
                              hipStream_t stream) {
  const float* x     = (const float*)d_in[0];
  const int*   ei    = (const int*)  d_in[1];
  const int*   et    = (const int*)  d_in[2];
  const int*   edges = (const int*)  d_in[3];
  const float* w1    = (const float*)d_in[4];
  const float* q1    = (const float*)d_in[5];
  const float* k1    = (const float*)d_in[6];
  const float* b1    = (const float*)d_in[7];
  const float* w2    = (const float*)d_in[8];
  const float* q2    = (const float*)d_in[9];
  const float* k2    = (const float*)d_in[10];
  const float* b2    = (const float*)d_in[11];
  const float* ln1g  = (const float*)d_in[12];
  const float* ln1b  = (const float*)d_in[13];
  const float* ln2g  = (const float*)d_in[14];
  const float* ln2b  = (const float*)d_in[15];
  const float* mw1   = (const float*)d_in[16];
  const float* mb1   = (const float*)d_in[17];
  const float* mw2   = (const float*)d_in[18];
  const float* mb2   = (const float*)d_in[19];
  float* out = (float*)d_out;

  // workspace layout (floats)
  float* ws   = (float*)d_ws;
  float* xw   = ws;                                   // N*R*HID = 51.2M
  float* aq   = xw   + (size_t)NN * RREL * HID;       // 1.6M
  float* ak   = aq   + (size_t)NN * RREL * HEADS;     // 1.6M
  float* ex   = ak   + (size_t)NN * RREL * HEADS;     // 2.56M
  float* amax = ex   + (size_t)EE * HEADS;            // 0.2M
  float* den  = amax + (size_t)NN * HEADS;            // 0.2M
  float* agg  = den  + (size_t)NN * HEADS;            // 6.4M
  float* h1   = agg  + (size_t)NN * HID;              // 6.4M
  float* h    = h1   + (size_t)NN * HID;              // 6.4M

  const dim3 blk(256);
  const int gInitAttn = (NN * HEADS + 255) / 256;
  const int gInitNode = NN * HID / 256;
  const dim3 gGemm(NN / MTB, RREL);
  const int gRowdot   = NN * RREL / 8;
  const int gEdgeH    = EE * HEADS / 256;
  const int gEdgeAgg  = EE * 32 / 256;
  const int gNode     = NN * 32 / 256;

  // ---------------- layer 1 ----------------
  k_init_attn<<<gInitAttn, blk, 0, stream>>>(amax, den);
  k_init_node<<<gInitNode, blk, 0, stream>>>(agg, b1);
  k_xw_gemm  <<<gGemm,     blk, 0, stream>>>(x, w1, xw);
  k_rowdot   <<<gRowdot,   blk, 0, stream>>>(xw, q1, k1, aq, ak);
  k_edge_alpha<<<gEdgeH,   blk, 0, stream>>>(aq, ak, ei, et, ex, amax);
  k_edge_exp <<<gEdgeH,    blk, 0, stream>>>(ex, amax, ei, den);
  k_edge_agg <<<gEdgeAgg,  blk, 0, stream>>>(xw, ex, den, ei, et, agg);
  k_postln   <<<gNode,     blk, 0, stream>>>(agg, ln1g, ln1b, nullptr, h1);

  // ---------------- layer 2 ----------------
  k_init_attn<<<gInitAttn, blk, 0, stream>>>(amax, den);
  k_init_node<<<gInitNode, blk, 0, stream>>>(agg, b2);
  k_xw_gemm  <<<gGemm,     blk, 0, stream>>>(h1, w2, xw);
  k_rowdot   <<<gRowdot,   blk, 0, stream>>>(xw, q2, k2, aq, ak);
  k_edge_alpha<<<gEdgeH,   blk, 0, stream>>>(aq, ak, ei, et, ex, amax);
  k_edge_exp <<<gEdgeH,    blk, 0, stream>>>(ex, amax, ei, den);
  k_edge_agg <<<gEdgeAgg,  blk, 0, stream>>>(xw, ex, den, ei, et, agg);
  k_postln   <<<gNode,     blk, 0, stream>>>(agg, ln2g, ln2b, h1, h);

  // ---------------- decode ----------------
  k_decode<<<EE2 / DECM, blk, 0, stream>>>(h, edges, mw1, mb1, mw2, mb2, out);
}